// NPTransitionPrior_25134148616668
// MI455X (gfx1250) — compile-verified
//
#include <hip/hip_runtime.h>
#include <math.h>

typedef __attribute__((ext_vector_type(2))) float v2f;
typedef __attribute__((ext_vector_type(8))) float v8f;

#define BB      64
#define TT      1002
#define DD      8
#define HH      64
#define FINF    17
#define NW      1000
#define NN      (BB * NW)            /* 64000  */
#define NTILES  (NN / 16)            /* 4000   */
#define W0STR   20                   /* FIN padded to 20 so K-chunks of 4 work */
#define WAVES   2
#define BLOCKS_PER_D 125

#define RES_OFF    0
#define LOGDET_OFF (NN * DD)         /* 512000 */
#define HIST_OFF   (LOGDET_OFF + BB) /* 512064 */

static __device__ __forceinline__ v8f zero8() {
  v8f z;
#pragma unroll
  for (int i = 0; i < 8; ++i) z[i] = 0.0f;
  return z;
}

/* D(16x16,f32) += A(16x4,f32) * B(4x16,f32) */
static __device__ __forceinline__ v8f wmma4(v2f a, v2f b, v8f c) {
  return __builtin_amdgcn_wmma_f32_16x16x4_f32(false, a, false, b, (short)0, c,
                                               false, false);
}

/* LDS ops are in-order per wave; this pins compiler ordering around the
   cross-lane staging round-trips and drains DScnt. */
static __device__ __forceinline__ void lds_fence() {
  asm volatile("s_wait_dscnt 0" ::: "memory");
}

__global__ __launch_bounds__(WAVES * 32)
void np_prior_kernel(const float* __restrict__ x,
                     const float* __restrict__ W0g, const float* __restrict__ b0g,
                     const float* __restrict__ a0g,
                     const float* __restrict__ W1g, const float* __restrict__ b1g,
                     const float* __restrict__ a1g,
                     const float* __restrict__ W2g, const float* __restrict__ b2g,
                     const float* __restrict__ a2g,
                     const float* __restrict__ W3g, const float* __restrict__ b3g,
                     float* __restrict__ res, float* __restrict__ logdet,
                     float* __restrict__ hist)
{
  __shared__ float W0s[HH * W0STR];          /* W0[h][f], f padded 17..19 = 0 */
  __shared__ float W1s[HH * HH];             /* W1[g][h] row-major            */
  __shared__ float W2s[HH * HH];
  __shared__ float W3s[HH];
  __shared__ float b0s[HH];
  __shared__ float b1s[HH];
  __shared__ float b2s[HH];
  __shared__ float stage[WAVES][3][16 * HH]; /* per-wave activation staging   */

  const int d   = blockIdx.y;
  const int tid = threadIdx.x;

  for (int i = tid; i < HH * W0STR; i += WAVES * 32) {
    const int h = i / W0STR;
    const int f = i - h * W0STR;
    W0s[i] = (f < FINF) ? W0g[(d * HH + h) * FINF + f] : 0.0f;
  }
  for (int i = tid; i < HH * HH; i += WAVES * 32) {
    W1s[i] = W1g[d * HH * HH + i];
    W2s[i] = W2g[d * HH * HH + i];
  }
  for (int i = tid; i < HH; i += WAVES * 32) {
    W3s[i] = W3g[d * HH + i];
    b0s[i] = b0g[d * HH + i];
    b1s[i] = b1g[d * HH + i];
    b2s[i] = b2g[d * HH + i];
  }
  const float a0v = a0g[d], a1v = a1g[d], a2v = a2g[d], b3v = b3g[d];
  __syncthreads();

  const int wave = tid >> 5;
  const int lane = tid & 31;
  const int m    = lane & 15;   /* column within 16-lane half       */
  const int hi   = lane >> 4;   /* which 16-lane half (0/1)         */

  float* S0 = &stage[wave][0][0];   /* h0 (forward), stays for dmask0 */
  float* S1 = &stage[wave][1][0];   /* h1 (forward) -> u (backward)   */
  float* SV = &stage[wave][2][0];   /* v  (backward) -> v2            */

  const int wid     = blockIdx.x * WAVES + wave;
  const int wstride = BLOCKS_PER_D * WAVES;

  for (int tile = wid; tile < NTILES; tile += wstride) {
    const int n0 = tile * 16;

    /* ---------- input fragments A0 (16 samples x 20 features, zero-padded) */
    const int n_m = n0 + m;
    const int bi  = n_m / NW;
    const int wi  = n_m - bi * NW;
    const float* xr = x + ((size_t)bi * TT + wi) * DD; /* 16 contig feats */

    v2f A0[5];
#pragma unroll
    for (int kc = 0; kc < 4; ++kc) {
      const int k0 = 4 * kc + 2 * hi;
      A0[kc].x = xr[k0];
      A0[kc].y = xr[k0 + 1];
    }
    {
      const float yl = xr[2 * DD + d];   /* x[b, w+2, d] = feature 16 */
      A0[4].x = hi ? 0.0f : yl;
      A0[4].y = 0.0f;
    }

    /* ---------- layer 0: z0 = inp @ W0^T ---------- */
    v8f C[4];
#pragma unroll
    for (int j = 0; j < 4; ++j) C[j] = zero8();
#pragma unroll
    for (int kc = 0; kc < 5; ++kc) {
      const int k0 = 4 * kc + 2 * hi;
#pragma unroll
      for (int j = 0; j < 4; ++j) {
        const v2f bf = *(const v2f*)(&W0s[(16 * j + m) * W0STR + k0]);
        C[j] = wmma4(A0[kc], bf, C[j]);
      }
    }
#pragma unroll
    for (int j = 0; j < 4; ++j) {
      const float bias = b0s[16 * j + m];
#pragma unroll
      for (int r = 0; r < 8; ++r) {
        const float z = C[j][r] + bias;
        S0[(r + 8 * hi) * HH + 16 * j + m] = (z > 0.0f) ? z : a0v * z;
      }
    }
    lds_fence();

    /* ---------- layer 1: z1 = h0 @ W1^T ---------- */
#pragma unroll
    for (int j = 0; j < 4; ++j) C[j] = zero8();
#pragma unroll 4
    for (int kc = 0; kc < 16; ++kc) {
      const int k0 = 4 * kc + 2 * hi;
      const v2f af = *(const v2f*)(&S0[m * HH + k0]);
#pragma unroll
      for (int j = 0; j < 4; ++j) {
        const v2f bf = *(const v2f*)(&W1s[(16 * j + m) * HH + k0]);
        C[j] = wmma4(af, bf, C[j]);
      }
    }
#pragma unroll
    for (int j = 0; j < 4; ++j) {
      const float bias = b1s[16 * j + m];
#pragma unroll
      for (int r = 0; r < 8; ++r) {
        const float z = C[j][r] + bias;
        S1[(r + 8 * hi) * HH + 16 * j + m] = (z > 0.0f) ? z : a1v * z;
      }
    }
    lds_fence();

    /* ---------- layer 2: z2 = h1 @ W2^T (keep h2 in regs) ---------- */
#pragma unroll
    for (int j = 0; j < 4; ++j) C[j] = zero8();
#pragma unroll 4
    for (int kc = 0; kc < 16; ++kc) {
      const int k0 = 4 * kc + 2 * hi;
      const v2f af = *(const v2f*)(&S1[m * HH + k0]);
#pragma unroll
      for (int j = 0; j < 4; ++j) {
        const v2f bf = *(const v2f*)(&W2s[(16 * j + m) * HH + k0]);
        C[j] = wmma4(af, bf, C[j]);
      }
    }
#pragma unroll
    for (int j = 0; j < 4; ++j) {
      const float bias = b2s[16 * j + m];
#pragma unroll
      for (int r = 0; r < 8; ++r) {
        const float z = C[j][r] + bias;
        C[j][r] = (z > 0.0f) ? z : a2v * z;   /* h2 */
      }
    }

    /* ---------- output head  +  v = W3 * d2 ---------- */
    float acc[8];
#pragma unroll
    for (int r = 0; r < 8; ++r) acc[r] = 0.0f;
#pragma unroll
    for (int j = 0; j < 4; ++j) {
      const float w3 = W3s[16 * j + m];
#pragma unroll
      for (int r = 0; r < 8; ++r) {
        const float h2v = C[j][r];
        acc[r] += h2v * w3;
        SV[(r + 8 * hi) * HH + 16 * j + m] = w3 * ((h2v > 0.0f) ? 1.0f : a2v);
      }
    }
    lds_fence();
#pragma unroll
    for (int msk = 1; msk < 16; msk <<= 1) {
#pragma unroll
      for (int r = 0; r < 8; ++r) acc[r] += __shfl_xor(acc[r], msk, 32);
    }
    if (m == 0) {
#pragma unroll
      for (int r = 0; r < 8; ++r) {
        const int n = n0 + r + 8 * hi;
        res[(size_t)n * DD + d] = acc[r] + b3v;
      }
    }

    /* ---------- backward: u = (v @ W2) * d1   (B = W2 natural) ---------- */
#pragma unroll
    for (int j = 0; j < 4; ++j) C[j] = zero8();
#pragma unroll 4
    for (int kc = 0; kc < 16; ++kc) {
      const int k0 = 4 * kc + 2 * hi;
      const v2f af = *(const v2f*)(&SV[m * HH + k0]);
#pragma unroll
      for (int j = 0; j < 4; ++j) {
        v2f bf;
        bf.x = W2s[(k0)     * HH + 16 * j + m];
        bf.y = W2s[(k0 + 1) * HH + 16 * j + m];
        C[j] = wmma4(af, bf, C[j]);
      }
    }
#pragma unroll
    for (int j = 0; j < 4; ++j) {
#pragma unroll
      for (int r = 0; r < 8; ++r) {
        const int idx = (r + 8 * hi) * HH + 16 * j + m;
        const float h1v = S1[idx];                   /* dmask1 from h1 sign */
        S1[idx] = C[j][r] * ((h1v > 0.0f) ? 1.0f : a1v);
      }
    }
    lds_fence();

    /* ---------- backward: v2 = (u @ W1) * d0   (B = W1 natural) ---------- */
#pragma unroll
    for (int j = 0; j < 4; ++j) C[j] = zero8();
#pragma unroll 4
    for (int kc = 0; kc < 16; ++kc) {
      const int k0 = 4 * kc + 2 * hi;
      const v2f af = *(const v2f*)(&S1[m * HH + k0]);
#pragma unroll
      for (int j = 0; j < 4; ++j) {
        v2f bf;
        bf.x = W1s[(k0)     * HH + 16 * j + m];
        bf.y = W1s[(k0 + 1) * HH + 16 * j + m];
        C[j] = wmma4(af, bf, C[j]);
      }
    }
#pragma unroll
    for (int j = 0; j < 4; ++j) {
#pragma unroll
      for (int r = 0; r < 8; ++r) {
        const int idx = (r + 8 * hi) * HH + 16 * j + m;
        const float h0v = S0[idx];                   /* dmask0 from h0 sign */
        C[j][r] *= ((h0v > 0.0f) ? 1.0f : a0v);
        SV[idx] = C[j][r];                           /* v2, kept in C too  */
      }
    }
    lds_fence();

    /* ---------- g[:,0:16] = v2 @ W0 (natural) -> hist_jac ---------- */
    v8f G = zero8();
#pragma unroll 4
    for (int kc = 0; kc < 16; ++kc) {
      const int k0 = 4 * kc + 2 * hi;
      const v2f af = *(const v2f*)(&SV[m * HH + k0]);
      v2f bf;
      bf.x = W0s[(k0)     * W0STR + m];
      bf.y = W0s[(k0 + 1) * W0STR + m];
      G = wmma4(af, bf, G);
    }
#pragma unroll
    for (int r = 0; r < 8; ++r) {
      const int n = n0 + r + 8 * hi;
      hist[((size_t)d * NN + n) * 16 + m] = G[r];
    }

    /* ---------- g[:,16] -> sum log|det| ---------- */
    float p[8];
#pragma unroll
    for (int r = 0; r < 8; ++r) p[r] = 0.0f;
#pragma unroll
    for (int j = 0; j < 4; ++j) {
      const float w016 = W0s[(16 * j + m) * W0STR + 16];
#pragma unroll
      for (int r = 0; r < 8; ++r) p[r] += C[j][r] * w016;
    }
#pragma unroll
    for (int msk = 1; msk < 16; msk <<= 1) {
#pragma unroll
      for (int r = 0; r < 8; ++r) p[r] += __shfl_xor(p[r], msk, 32);
    }
    if (m == 0) {
#pragma unroll
      for (int r = 0; r < 8; ++r) {
        const int n    = n0 + r + 8 * hi;
        const int bidx = n / NW;
        atomicAdd(&logdet[bidx], logf(fabsf(p[r])));
      }
    }
  }
}

__global__ void zero_logdet_kernel(float* __restrict__ p) {
  if (threadIdx.x < BB) p[threadIdx.x] = 0.0f;
}

extern "C" void kernel_launch(void* const* d_in, const int* in_sizes, int n_in,
                              void* d_out, int out_size, void* d_ws, size_t ws_size,
                              hipStream_t stream) {
  (void)in_sizes; (void)n_in; (void)out_size; (void)d_ws; (void)ws_size;
  const float* x  = (const float*)d_in[0];
  const float* W0 = (const float*)d_in[1];
  const float* b0 = (const float*)d_in[2];
  const float* a0 = (const float*)d_in[3];
  const float* W1 = (const float*)d_in[4];
  const float* b1 = (const float*)d_in[5];
  const float* a1 = (const float*)d_in[6];
  const float* W2 = (const float*)d_in[7];
  const float* b2 = (const float*)d_in[8];
  const float* a2 = (const float*)d_in[9];
  const float* W3 = (const float*)d_in[10];
  const float* b3 = (const float*)d_in[11];

  float* out    = (float*)d_out;
  float* res    = out + RES_OFF;
  float* logdet = out + LOGDET_OFF;
  float* hist   = out + HIST_OFF;

  zero_logdet_kernel<<<1, 64, 0, stream>>>(logdet);

  dim3 grid(BLOCKS_PER_D, DD, 1);
  np_prior_kernel<<<grid, WAVES * 32, 0, stream>>>(
      x, W0, b0, a0, W1, b1, a1, W2, b2, a2, W3, b3, res, logdet, hist);
}